// LinearRNNCell_72765335929666
// MI455X (gfx1250) — compile-verified
//
#include <hip/hip_runtime.h>
#include <stdint.h>

typedef float v2f __attribute__((ext_vector_type(2)));
typedef float v4f __attribute__((ext_vector_type(4)));
typedef float v8f __attribute__((ext_vector_type(8)));
typedef unsigned int u32x4 __attribute__((ext_vector_type(4)));
typedef int i32x8 __attribute__((ext_vector_type(8)));
typedef int i32x4 __attribute__((ext_vector_type(4)));

#define T_STEPS 2048
#define BATCH   16
#define IN      512
#define HID     512
#define WROW    1024               // weight row stride = I + H
#define LDS_STRIDE 516             // padded LDS row stride (floats)
#define PACK_MAT_FLOATS (32 * 64 * 32 * 4)   // one packed 512x512 matrix = 1 MB

// ---------------------------------------------------------------------------
// Prologue: repack w_hh and w_xh into WMMA-fragment order.
// Layout: [matrix(2)][tile(32)][kp(64)][lane(32)][float4]
//   lane (lr=lane&15, lh=lane>>4) holds B[k][n] for n = 16*tile+lr,
//   k = 8*kp + 2*lh + {0,1} (.xy) and 8*kp + 4 + 2*lh + {0,1} (.zw).
// Consumers then stream 512 B contiguous per wave per K-pair (global_load_b128).
// ---------------------------------------------------------------------------
__global__ __launch_bounds__(256) void pack_weights_kernel(
    const float* __restrict__ weight, float* __restrict__ ws)
{
  const int idx  = blockIdx.x * 256 + threadIdx.x;   // 0 .. 131071
  const int mat  = idx >> 16;                        // 0 = w_hh, 1 = w_xh
  const int tile = (idx >> 11) & 31;
  const int kp   = (idx >> 5) & 63;
  const int lane = idx & 31;
  const int lr = lane & 15, lh = lane >> 4;
  const int n  = tile * 16 + lr;
  const int kb = kp * 8 + 2 * lh;
  const float* src = weight + (size_t)n * WROW + (mat ? HID : 0) + kb;
  v4f v = { src[0], src[1], src[4], src[5] };
  *(v4f*)(ws + (size_t)idx * 4) = v;
}

// ---------------------------------------------------------------------------
// TDM: DMA a 64x512 f32 tile (row stride 512) into LDS with 2-dword padding
// every 256 dwords -> LDS row stride 516 floats, conflict-free ds_load_b64.
// ---------------------------------------------------------------------------
__device__ inline void tdm_load_64x512(const float* gsrc, uint32_t lds_byte_off) {
  const uint64_t ga = (uint64_t)(uintptr_t)gsrc;
  u32x4 g0;
  g0[0] = 1u;                                             // count = 1 descriptor
  g0[1] = lds_byte_off;                                   // lds_addr
  g0[2] = (uint32_t)ga;                                   // global_addr[31:0]
  g0[3] = (uint32_t)((ga >> 32) & 0x1FFFFFFull)           // global_addr[56:32]
        | (2u << 30);                                     // type = 2 (image)
  i32x8 g1;
  g1[0] = (int)((2u << 16)      // data_size = 4 bytes
              | (1u << 20)      // pad_enable
              | (7u << 22)      // pad_interval: 256 dwords
              | (1u << 25));    // pad_amount: 2 dwords  -> stride 516, 8B-aligned
  g1[1] = (int)(512u << 16);    // tensor_dim0 = 512 (low 16 bits)
  g1[2] = (int)(0x8000u << 16); // tensor_dim1 = 32768 (low 16 bits)
  g1[3] = (int)(512u << 16);    // tile_dim0 = 512
  g1[4] = 64;                   // tile_dim1 = 64 rows, tile_dim2 = 0 (2D)
  g1[5] = 512;                  // tensor_dim0_stride = 512 (low 32)
  g1[6] = 0;
  g1[7] = 0;
  const i32x4 z4 = {0, 0, 0, 0};                          // groups 2/3 unused (2D)
  const i32x8 z8 = {0, 0, 0, 0, 0, 0, 0, 0};
  __builtin_amdgcn_tensor_load_to_lds(g0, g1, z4, z4, z8, 0);
  __builtin_amdgcn_s_wait_tensorcnt(0);
}

// ---------------------------------------------------------------------------
// Kernel A: x_proj = inputs @ w_xh^T + bias  (M=32768, N=512, K=512)
// Block tile 64x512: A strip staged by TDM into LDS (132 KB), 8 waves each
// compute 4 M-tiles x 4 N-tiles with f32 WMMA; B streamed coalesced (packed).
// ---------------------------------------------------------------------------
__global__ __launch_bounds__(256) void xproj_wmma_kernel(
    const float* __restrict__ inp,       // (32768, 512)
    const float* __restrict__ wxh_pack,  // packed w_xh fragments
    const float* __restrict__ bias,      // (512)
    float* __restrict__ out)             // (32768, 512)
{
  extern __shared__ float atile[];       // 64 * 516 floats = 132096 B
  const int lane = threadIdx.x & 31;
  const int wave = threadIdx.x >> 5;
  const int lr = lane & 15, lh = lane >> 4;
  const int m0 = blockIdx.x * 64;

  if (wave == 0) {
    tdm_load_64x512(inp + (size_t)m0 * IN, (uint32_t)(size_t)atile);
  }
  __syncthreads();

  // Packed B streams: one per N-tile of this wave (512 B / K-pair, coalesced).
  const float* bp0 = wxh_pack + (size_t)(wave * 4 + 0) * 8192 + lane * 4;
  const float* bp1 = wxh_pack + (size_t)(wave * 4 + 1) * 8192 + lane * 4;
  const float* bp2 = wxh_pack + (size_t)(wave * 4 + 2) * 8192 + lane * 4;
  const float* bp3 = wxh_pack + (size_t)(wave * 4 + 3) * 8192 + lane * 4;
  // A fragment base in LDS (row lr, K pair at 2*lh)
  const float* al = atile + lr * LDS_STRIDE + 2 * lh;

  v8f acc[4][4] = {};

  for (int kp = 0; kp < 64; ++kp) {       // K-pair loop: K = 8*kp .. 8*kp+7
    const int kofs = kp * 8 + 2 * (kp >> 5);    // + TDM pad adjustment (even)
    v2f a0[4], a1[4];
#pragma unroll
    for (int mt = 0; mt < 4; ++mt) {
      const float* am = al + mt * (16 * LDS_STRIDE);
      a0[mt] = *(const v2f*)(am + kofs);
      a1[mt] = *(const v2f*)(am + kofs + 4);
    }
    v4f wv[4];
    wv[0] = *(const v4f*)(bp0 + kp * 128);
    wv[1] = *(const v4f*)(bp1 + kp * 128);
    wv[2] = *(const v4f*)(bp2 + kp * 128);
    wv[3] = *(const v4f*)(bp3 + kp * 128);
#pragma unroll
    for (int nt = 0; nt < 4; ++nt) {
      v2f w0 = { wv[nt].x, wv[nt].y };
      v2f w1 = { wv[nt].z, wv[nt].w };
#pragma unroll
      for (int mt = 0; mt < 4; ++mt) {
        acc[mt][nt] = __builtin_amdgcn_wmma_f32_16x16x4_f32(
            false, a0[mt], false, w0, (short)0, acc[mt][nt], false, false);
        acc[mt][nt] = __builtin_amdgcn_wmma_f32_16x16x4_f32(
            false, a1[mt], false, w1, (short)0, acc[mt][nt], false, false);
      }
    }
  }

  // C/D layout: VGPR r, lane l -> element (r + 8*(l>=16), l&15)
#pragma unroll
  for (int nt = 0; nt < 4; ++nt) {
    const int n = wave * 64 + nt * 16 + lr;
    const float bb = bias[n];
#pragma unroll
    for (int mt = 0; mt < 4; ++mt) {
#pragma unroll
      for (int r = 0; r < 8; ++r) {
        const int m = m0 + mt * 16 + r + lh * 8;
        out[(size_t)m * HID + n] = acc[mt][nt][r] + bb;
      }
    }
  }
}

// ---------------------------------------------------------------------------
// Kernel B: sequential scan  h_t = h_{t-1} @ w_hh^T + xp_t
// One workgroup (32 wave32s, one WGP). h in LDS (stride 516, conflict-free);
// packed w_hh streamed with coalesced b128 (one load covers two K-steps).
// outputs[t] read (xp_t) and overwritten (h_t) in place in d_out.
// ---------------------------------------------------------------------------
__global__ __launch_bounds__(1024) void rnn_scan_wmma_kernel(
    const float* __restrict__ whh_pack,  // packed w_hh fragments
    const float* __restrict__ state,     // (16, 512)
    float* __restrict__ out)             // [T*B*H] outputs, then [B*H] last
{
  __shared__ float hbuf[BATCH * LDS_STRIDE];   // ~33 KB

  const int tid  = threadIdx.x;
  const int lane = tid & 31;
  const int wave = tid >> 5;            // 0..31 -> column strip [16w, 16w+16)
  const int lr   = lane & 15;
  const int lh   = lane >> 4;
  const int n0   = wave * 16;

  for (int idx = tid; idx < BATCH * HID; idx += 1024) {
    hbuf[(idx >> 9) * LDS_STRIDE + (idx & 511)] = state[idx];
  }
  __syncthreads();

  const float* wp   = whh_pack + (size_t)wave * 8192 + lane * 4;
  const float* aldz = hbuf + lr * LDS_STRIDE + 2 * lh;

  for (int t = 0; t < T_STEPS; ++t) {
    float* xrow = out + (size_t)t * (BATCH * HID);

    float xp[8];
#pragma unroll
    for (int r = 0; r < 8; ++r) {
      xp[r] = xrow[(size_t)(r + lh * 8) * HID + n0 + lr];
    }

    v8f accA = {}, accB = {};
#pragma unroll 4
    for (int kp = 0; kp < 64; ++kp) {
      v4f wv = *(const v4f*)(wp + kp * 128);
      v2f a0 = *(const v2f*)(aldz + kp * 8);
      v2f a1 = *(const v2f*)(aldz + kp * 8 + 4);
      v2f w0 = { wv.x, wv.y };
      v2f w1 = { wv.z, wv.w };
      accA = __builtin_amdgcn_wmma_f32_16x16x4_f32(false, a0, false, w0, (short)0, accA, false, false);
      accB = __builtin_amdgcn_wmma_f32_16x16x4_f32(false, a1, false, w1, (short)0, accB, false, false);
    }
    v8f acc = accA + accB;

    __syncthreads();                    // all reads of h_{t-1} complete

#pragma unroll
    for (int r = 0; r < 8; ++r) {
      const int m = r + lh * 8;
      const float v = acc[r] + xp[r];
      xrow[(size_t)m * HID + n0 + lr] = v;      // outputs[t]
      hbuf[m * LDS_STRIDE + n0 + lr]  = v;      // h_t
    }
    __syncthreads();                    // h_t visible to all waves
  }

  float* lastp = out + (size_t)T_STEPS * (BATCH * HID);
  for (int idx = tid; idx < BATCH * HID; idx += 1024) {
    lastp[idx] = hbuf[(idx >> 9) * LDS_STRIDE + (idx & 511)];
  }
}

// ---------------------------------------------------------------------------
extern "C" void kernel_launch(void* const* d_in, const int* in_sizes, int n_in,
                              void* d_out, int out_size, void* d_ws, size_t ws_size,
                              hipStream_t stream) {
  const float* inp    = (const float*)d_in[0];  // (2048, 16, 512)
  const float* state  = (const float*)d_in[1];  // (16, 512)
  const float* weight = (const float*)d_in[2];  // (512, 1024)
  const float* bias   = (const float*)d_in[3];  // (512)
  float* out = (float*)d_out;
  float* ws  = (float*)d_ws;                    // needs 2 MB
  float* whh_pack = ws;
  float* wxh_pack = ws + PACK_MAT_FLOATS;

  pack_weights_kernel<<<dim3(512), dim3(256), 0, stream>>>(weight, ws);

  const int xproj_lds = 64 * LDS_STRIDE * 4;    // 132096 B (CDNA5: 320 KB/WGP)
  (void)hipFuncSetAttribute((const void*)xproj_wmma_kernel,
                            hipFuncAttributeMaxDynamicSharedMemorySize, xproj_lds);
  xproj_wmma_kernel<<<dim3((T_STEPS * BATCH) / 64), dim3(256), xproj_lds, stream>>>(
      inp, wxh_pack, bias, out);

  rnn_scan_wmma_kernel<<<dim3(1), dim3(1024), 0, stream>>>(whh_pack, state, out);
}